// InterferenceLayer_46686294507821
// MI455X (gfx1250) — compile-verified
//
#include <hip/hip_runtime.h>
#include <hip/hip_bf16.h>

typedef __attribute__((ext_vector_type(2))) float v2f;
typedef __attribute__((ext_vector_type(8))) float v8f;
typedef __attribute__((ext_vector_type(4))) int   v4i;

// D = A(16x4 f32) * B(4x16 f32) + C(16x16 f32), wave32 WMMA
static __device__ __forceinline__ v8f wmma_k4(v2f a, v2f b, v8f c) {
    return __builtin_amdgcn_wmma_f32_16x16x4_f32(
        /*neg_a=*/false, a, /*neg_b=*/false, b,
        /*c_mod=*/(short)0, c, /*reuse_a=*/false, /*reuse_b=*/false);
}

// ---- gfx1250 async global->LDS staging (ASYNCcnt-tracked) ------------------
#if defined(__gfx1250__) && __has_builtin(__builtin_amdgcn_global_load_async_to_lds_b128)
#define USE_ASYNC_LDS 1
#else
#define USE_ASYNC_LDS 0
#endif

typedef __attribute__((address_space(1))) v4i g_v4i;   // global int4
typedef __attribute__((address_space(3))) v4i l_v4i;   // LDS int4
static __device__ __forceinline__ g_v4i* as_global_v4i(const void* p) {
    return (g_v4i*)(uintptr_t)p;
}
static __device__ __forceinline__ l_v4i* as_lds_v4i(const void* p) {
    return (l_v4i*)(unsigned)(uintptr_t)p;   // low 32 bits = LDS byte offset
}
template <int N> static __device__ __forceinline__ void wait_asynccnt() {
#if __has_builtin(__builtin_amdgcn_s_wait_asynccnt)
    __builtin_amdgcn_s_wait_asynccnt(N);
#else
    asm volatile("s_wait_asynccnt %0" ::"i"(N) : "memory");
#endif
}

// ---------------------------------------------------------------------------
// Kernel 0: fold conv weights through the projection.
// W2[96][192] = proj_w[96][32] @ conv_w[32][192];  b2 = proj_w@conv_b + proj_b
// ---------------------------------------------------------------------------
__global__ void fuse_weights_kernel(const float* __restrict__ conv_w,
                                    const float* __restrict__ conv_b,
                                    const float* __restrict__ proj_w,
                                    const float* __restrict__ proj_b,
                                    float* __restrict__ W2,
                                    float* __restrict__ b2) {
    const int t = threadIdx.x;            // 0..95
    const float* pr = proj_w + t * 32;
    for (int e = 0; e < 192; ++e) {
        float s = 0.f;
        #pragma unroll
        for (int m = 0; m < 32; ++m) s += pr[m] * conv_w[m * 192 + e];
        W2[t * 192 + e] = s;
    }
    float sb = proj_b[t];
    #pragma unroll
    for (int m = 0; m < 32; ++m) sb += pr[m] * conv_b[m];
    b2[t] = sb;
}

// ---------------------------------------------------------------------------
// Kernel 1: patch embed + qkv projection in one pass.
// One block (96 threads) per token; patch (192 floats) staged in LDS.
// V is written TRANSPOSED (vt[b][32][4096]) so the attention kernel's staged
// V tiles are [d][n] and every WMMA B-fragment is a contiguous b64.
// ---------------------------------------------------------------------------
__global__ void embed_kernel(const float* __restrict__ x,
                             const float* __restrict__ W2,
                             const float* __restrict__ b2,
                             float* __restrict__ q,
                             float* __restrict__ k,
                             float* __restrict__ vt) {
    const int token = blockIdx.x;         // 0..16383
    const int b = token >> 12;            // 4096 tokens per batch
    const int n = token & 4095;
    const int h = n >> 6, w = n & 63;
    __shared__ float patch[192];
    const int t = threadIdx.x;            // 0..95
    for (int e = t; e < 192; e += 96) {
        const int c = e >> 6, rem = e & 63, p = rem >> 3, qq = rem & 7;
        patch[e] = x[(((size_t)b * 3 + c) * 512 + (h * 8 + p)) * 512 + (w * 8 + qq)];
    }
    __syncthreads();
    const float* wr = W2 + t * 192;
    float acc = b2[t];
    #pragma unroll 8
    for (int e = 0; e < 192; ++e) acc += patch[e] * wr[e];
    const int d = t & 31;
    if (t < 32)      q[((size_t)b * 4096 + n) * 32 + d] = acc;
    else if (t < 64) k[((size_t)b * 4096 + n) * 32 + d] = acc;
    else             vt[((size_t)b * 32 + d) * 4096 + n] = acc;
}

// ---------------------------------------------------------------------------
// Kernel 2: fused attention  out = cos(2*pi * (Q K^T, zero diag)) @ V
// 4 waves/block, 1 Q row-tile per wave. K/V 16-key tiles are staged into LDS
// cooperatively with double buffering via global_load_async_to_lds_b128
// (ASYNCcnt), so the next tile's fill overlaps the current tile's 16 fp32
// WMMAs. All fragments in the hot loop are ds_load_b64.
// ---------------------------------------------------------------------------
__global__ void __launch_bounds__(128)
attn_kernel(const float* __restrict__ q,
            const float* __restrict__ k,
            const float* __restrict__ vt,
            float* __restrict__ out) {
    __shared__ float ldsK[2][16 * 32];    // [buf][key n][dim d]
    __shared__ float ldsV[2][32 * 16];    // [buf][dim d][key n]
    __shared__ float ptile[4][16 * 16];   // per-wave P tile (C->A relayout)

    const int t    = threadIdx.x;         // 0..127
    const int lane = t & 31;
    const int wave = t >> 5;
    const int b    = blockIdx.x >> 6;     // 64 blocks per batch
    const int i    = (blockIdx.x & 63) * 4 + wave;  // Q row-tile index
    const int hi   = lane >> 4;
    const int l16  = lane & 15;

    const float* qb  = q  + ((size_t)b * 4096 + i * 16) * 32;
    const float* kb  = k  + (size_t)b * 4096 * 32;
    const float* vtb = vt + (size_t)b * 32 * 4096;
    float* pt = ptile[wave];

    // staging addressing: both tiles are a flat 128 x 16B copy
    const int krow = t >> 3, kcol = (t & 7) * 4;    // K tile: 16 x 32
    const int vrow = t >> 2, vcol = (t & 3) * 4;    // V tile: 32 x 16

    // Q fragment, A-layout: lane row=l16, VGPR pair = K {c*4+2*hi, +1}
    v2f a_q[8];
    #pragma unroll
    for (int c = 0; c < 8; ++c)
        a_q[c] = *(const v2f*)(qb + l16 * 32 + c * 4 + hi * 2);

    v8f acc0 = {}; v8f acc1 = {};         // D tile: dims 0-15 / 16-31

    // ---- stage tile j into buffer buf ----
    auto stage = [&](int j, int buf) {
        const float* gk = kb  + ((size_t)(j * 16 + krow)) * 32 + kcol;
        const float* gv = vtb + (size_t)vrow * 4096 + j * 16 + vcol;
#if USE_ASYNC_LDS
        __builtin_amdgcn_global_load_async_to_lds_b128(
            as_global_v4i(gk), as_lds_v4i(&ldsK[buf][t * 4]), 0, 0);
        __builtin_amdgcn_global_load_async_to_lds_b128(
            as_global_v4i(gv), as_lds_v4i(&ldsV[buf][t * 4]), 0, 0);
#else
        *(float4*)(&ldsK[buf][t * 4]) = *(const float4*)gk;
        *(float4*)(&ldsV[buf][t * 4]) = *(const float4*)gv;
#endif
    };

    stage(0, 0);
    for (int j = 0; j < 256; ++j) {
        const int cur = j & 1;
        if (j + 1 < 256) {
            stage(j + 1, cur ^ 1);
#if USE_ASYNC_LDS
            wait_asynccnt<2>();           // drain stage j, let stage j+1 fly
#endif
        } else {
#if USE_ASYNC_LDS
            wait_asynccnt<0>();
#endif
        }
        __syncthreads();

        const float* kc = ldsK[cur];
        const float* vc = ldsV[cur];

        // ---- S = Qtile * Ktile^T (16x16), K-dim = 32 in 8 WMMA steps ----
        v8f s = {};
        #pragma unroll
        for (int c = 0; c < 8; ++c) {
            v2f bk = *(const v2f*)(kc + l16 * 32 + c * 4 + hi * 2);  // ds_load_b64
            s = wmma_k4(a_q[c], bk, s);
        }
        // ---- P = cos(2*pi*S) (v_cos_f32 semantics), diag forced to 1 ----
        #pragma unroll
        for (int r = 0; r < 8; ++r) {
            const int m = r + hi * 8;
            float pv = __builtin_amdgcn_cosf(s[r]);
            pv = (i == j && m == l16) ? 1.0f : pv;
            pt[m * 16 + l16] = pv;
        }
        // ---- D += Ptile(16x16) * Vtile(16x32), inner dim 16 in 4 steps ----
        #pragma unroll
        for (int c = 0; c < 4; ++c) {
            v2f ap  = *(const v2f*)(pt + l16 * 16 + c * 4 + hi * 2);         // A-frag
            v2f bv0 = *(const v2f*)(vc + l16 * 16 + c * 4 + hi * 2);         // d 0-15
            acc0 = wmma_k4(ap, bv0, acc0);
            v2f bv1 = *(const v2f*)(vc + (l16 + 16) * 16 + c * 4 + hi * 2);  // d 16-31
            acc1 = wmma_k4(ap, bv1, acc1);
        }
        __syncthreads();                  // all reads done before buf reuse
    }

    // ---- write D tile: out[b][i*16+m][d], C-layout scatter ----
    float* ob = out + ((size_t)b * 4096 + i * 16) * 32;
    #pragma unroll
    for (int r = 0; r < 8; ++r) {
        const int m = r + hi * 8;
        ob[m * 32 + l16]      = acc0[r];
        ob[m * 32 + 16 + l16] = acc1[r];
    }
}

// ---------------------------------------------------------------------------
// Kernel 3: ConvTranspose2d with kernel=stride=8 (pure per-pixel epilogue).
// ---------------------------------------------------------------------------
__global__ void deconv_kernel(const float* __restrict__ inter,
                              const float* __restrict__ dw,
                              const float* __restrict__ db,
                              float* __restrict__ out, int total) {
    const int idx = blockIdx.x * 256 + threadIdx.x;
    if (idx >= total) return;
    const int xx = idx & 511;
    const int yy = (idx >> 9) & 511;
    const int bo = idx >> 18;             // b*3 + o
    const int o  = bo % 3;
    const int b  = bo / 3;
    const int h = yy >> 3, p = yy & 7, w = xx >> 3, qq = xx & 7;
    const float* ib = inter + ((size_t)b * 4096 + (h * 64 + w)) * 32;
    const int woff = o * 64 + p * 8 + qq;
    float acc = db[o];
    #pragma unroll
    for (int d = 0; d < 32; ++d) acc += ib[d] * dw[d * 192 + woff];
    out[idx] = acc;
}

// ---------------------------------------------------------------------------
extern "C" void kernel_launch(void* const* d_in, const int* in_sizes, int n_in,
                              void* d_out, int out_size, void* d_ws, size_t ws_size,
                              hipStream_t stream) {
    const float* x        = (const float*)d_in[0];
    const float* conv_w   = (const float*)d_in[1];
    const float* conv_b   = (const float*)d_in[2];
    const float* proj_w   = (const float*)d_in[3];
    const float* proj_b   = (const float*)d_in[4];
    const float* deconv_w = (const float*)d_in[5];
    const float* deconv_b = (const float*)d_in[6];
    float* out = (float*)d_out;

    float* ws   = (float*)d_ws;
    float* W2   = ws;                      // 96*192  = 18432 floats
    float* b2   = W2 + 18432;              // 96 (padded to 128)
    float* qbuf = b2 + 128;                // 4*4096*32 = 524288 floats
    float* kbuf = qbuf + 524288;
    float* vtbf = kbuf + 524288;           // V transposed [b][32][4096]
    float* ivec = vtbf + 524288;           // inter: 524288 floats

    fuse_weights_kernel<<<1, 96, 0, stream>>>(conv_w, conv_b, proj_w, proj_b, W2, b2);
    embed_kernel<<<16384, 96, 0, stream>>>(x, W2, b2, qbuf, kbuf, vtbf);
    attn_kernel<<<256, 128, 0, stream>>>(qbuf, kbuf, vtbf, ivec);
    deconv_kernel<<<(3145728 + 255) / 256, 256, 0, stream>>>(ivec, deconv_w, deconv_b, out, 3145728);
}